// NGPDensityField_1726576856125
// MI455X (gfx1250) — compile-verified
//
#include <hip/hip_runtime.h>
#include <math.h>

typedef __attribute__((ext_vector_type(2))) float v2f;
typedef __attribute__((ext_vector_type(8))) float v8f;

#define TBL_SIZE 131072u   // T = 2^17 entries per level
#define BLOCK 256

// Multires hash-grid encode + 2-layer MLP density head, one point per thread,
// MLP evaluated with f32 WMMA (16x16x4) per wave over 16-point row tiles.
__global__ __launch_bounds__(BLOCK) void ngp_density_kernel(
    const float* __restrict__ positions,   // [N,3]
    const float* __restrict__ table,       // [5, 2^17, 2]
    const float* __restrict__ W1,          // [10, 64]
    const float* __restrict__ W2,          // [64, 1]
    const float* __restrict__ aabb,        // [6]
    float* __restrict__ out,               // [N, 1]
    int N)
{
    // enc rows: stride 12 floats (10 features + 2 zero pad for K=12 WMMA padding)
    __shared__ __align__(16) float encLds[BLOCK * 12];
    __shared__ float selLds[BLOCK];

    const int tid = threadIdx.x;
    const int pt  = blockIdx.x * BLOCK + tid;
    const int ptc = (pt < N) ? pt : (N - 1);   // clamp loads; keep EXEC full for WMMA

    // ---- normalize into [0,1] box ----
    const float a0 = aabb[0], a1 = aabb[1], a2 = aabb[2];
    const float i0 = 1.f / (aabb[3] - a0);
    const float i1 = 1.f / (aabb[4] - a1);
    const float i2 = 1.f / (aabb[5] - a2);
    const float px = (positions[ptc * 3 + 0] - a0) * i0;
    const float py = (positions[ptc * 3 + 1] - a1) * i1;
    const float pz = (positions[ptc * 3 + 2] - a2) * i2;

    const bool inside = (px > 0.f) & (px < 1.f) & (py > 0.f) & (py < 1.f) &
                        (pz > 0.f) & (pz < 1.f);
    selLds[tid] = inside ? 1.f : 0.f;

    // per-level scales (f32 of BASE*PLS^l - 1) and resolutions (ceil(scale)+1)
    const float    scales[5] = {15.f, 25.908685288118864f, 44.254833995939045f,
                                75.10925536017414f, 127.f};
    const uint32_t resv[5]   = {16u, 27u, 46u, 77u, 128u};

    const float* encRow = &encLds[tid * 12];
#pragma unroll
    for (int l = 0; l < 5; ++l) {
        const float s = scales[l];
        const float plx = fmaf(px, s, 0.5f);
        const float ply = fmaf(py, s, 0.5f);
        const float plz = fmaf(pz, s, 0.5f);
        const float fx = floorf(plx), fy = floorf(ply), fz = floorf(plz);
        const float wx = plx - fx, wy = ply - fy, wz = plz - fz;
        const uint32_t ux = (uint32_t)fx, uy = (uint32_t)fy, uz = (uint32_t)fz;

        const v2f* __restrict__ tbl = (const v2f*)table + (size_t)l * TBL_SIZE;
        float acc0 = 0.f, acc1 = 0.f;
#pragma unroll
        for (int c = 0; c < 8; ++c) {
            const uint32_t cx = ux + (c & 1);
            const uint32_t cy = uy + ((c >> 1) & 1);
            const uint32_t cz = uz + (c >> 2);
            uint32_t idx;
            if (l < 3) {  // dense levels: res^3 <= T
                idx = cx + cy * resv[l] + cz * (resv[l] * resv[l]);
            } else {      // hashed levels
                idx = (cx ^ (cy * 2654435761u) ^ (cz * 805459861u)) & (TBL_SIZE - 1u);
            }
            const float wgt = ((c & 1)        ? wx : 1.f - wx) *
                              (((c >> 1) & 1) ? wy : 1.f - wy) *
                              ((c >> 2)       ? wz : 1.f - wz);
            const v2f t = tbl[idx];
            acc0 = fmaf(wgt, t.x, acc0);
            acc1 = fmaf(wgt, t.y, acc1);
        }
        encLds[tid * 12 + 2 * l + 0] = acc0;
        encLds[tid * 12 + 2 * l + 1] = acc1;
    }
    encLds[tid * 12 + 10] = 0.f;   // K padding (10,11) for K=12 WMMA sweep
    encLds[tid * 12 + 11] = 0.f;
    (void)encRow;

    __syncthreads();

    // ---- MLP: enc[16,10(->12)] @ W1[10,64] via V_WMMA_F32_16X16X4_F32 ----
    const int lane = tid & 31;
    const int wv   = tid >> 5;
    const int hl   = lane >> 4;    // half-wave select (K+2 / M+8 side)
    const int c16  = lane & 15;

    // B fragments: W1 columns, 3 K-steps x 4 N-tiles (K slots >=10 are zero)
    v2f bfrag[4][3];
    float w2v[4];
#pragma unroll
    for (int nt = 0; nt < 4; ++nt) {
        const int col = nt * 16 + c16;
#pragma unroll
        for (int ks = 0; ks < 3; ++ks) {
            const int k = ks * 4 + hl * 2;
            bfrag[nt][ks].x = (k     < 10) ? W1[k * 64 + col]       : 0.f;
            bfrag[nt][ks].y = (k + 1 < 10) ? W1[(k + 1) * 64 + col] : 0.f;
        }
        w2v[nt] = W2[col];
    }

#pragma unroll
    for (int mt = 0; mt < 2; ++mt) {
        const int rowbase = wv * 32 + mt * 16;   // wave-local 16-row tile

        // A fragments from LDS (aligned float2: stride 12, even k)
        v2f afrag[3];
#pragma unroll
        for (int ks = 0; ks < 3; ++ks) {
            const int k = ks * 4 + hl * 2;
            afrag[ks] = *(const v2f*)&encLds[(rowbase + c16) * 12 + k];
        }

        float partial[8] = {0.f, 0.f, 0.f, 0.f, 0.f, 0.f, 0.f, 0.f};
#pragma unroll
        for (int nt = 0; nt < 4; ++nt) {
            v8f c = {0.f, 0.f, 0.f, 0.f, 0.f, 0.f, 0.f, 0.f};
#pragma unroll
            for (int ks = 0; ks < 3; ++ks) {
                c = __builtin_amdgcn_wmma_f32_16x16x4_f32(
                        false, afrag[ks], false, bfrag[nt][ks],
                        (short)0, c, false, false);
            }
            // layer 2 (64->1): relu then dot with W2 slice, from C layout
            // C[r] @ lane: M = r + 8*hl, N = nt*16 + c16
#pragma unroll
            for (int r = 0; r < 8; ++r)
                partial[r] = fmaf(fmaxf(c[r], 0.f), w2v[nt], partial[r]);
        }

        // butterfly-reduce the 16 N-lanes of each half-wave
#pragma unroll
        for (int mask = 1; mask <= 8; mask <<= 1) {
#pragma unroll
            for (int r = 0; r < 8; ++r)
                partial[r] += __shfl_xor(partial[r], mask, 32);
        }

        // lane c16 (<8) of each half writes row rowbase + 8*hl + c16
        float db = partial[0];
#pragma unroll
        for (int r = 1; r < 8; ++r) db = (c16 == r) ? partial[r] : db;

        if (c16 < 8) {
            const int row  = rowbase + hl * 8 + c16;
            const int opt  = blockIdx.x * BLOCK + row;
            if (opt < N)
                out[opt] = expf(db - 1.f) * selLds[row];
        }
    }
}

extern "C" void kernel_launch(void* const* d_in, const int* in_sizes, int n_in,
                              void* d_out, int out_size, void* d_ws, size_t ws_size,
                              hipStream_t stream) {
    const float* positions = (const float*)d_in[0];
    const float* table     = (const float*)d_in[1];
    const float* W1        = (const float*)d_in[2];
    const float* W2        = (const float*)d_in[3];
    const float* aabb      = (const float*)d_in[4];
    float* out = (float*)d_out;

    const int N = in_sizes[0] / 3;
    const int blocks = (N + BLOCK - 1) / BLOCK;
    hipLaunchKernelGGL(ngp_density_kernel, dim3(blocks), dim3(BLOCK), 0, stream,
                       positions, table, W1, W2, aabb, out, N);
}